// SimpleMACE_2362232013054
// MI455X (gfx1250) — compile-verified
//
#include <hip/hip_runtime.h>
#include <hip/hip_bf16.h>
#include <math.h>

typedef float v2f __attribute__((ext_vector_type(2)));
typedef float v8f __attribute__((ext_vector_type(8)));

// fp32 matrix path on CDNA5: V_WMMA_F32_16X16X4_F32 (16x16 f32 D/C, 16x4 A, 4x16 B)
__device__ __forceinline__ v8f wmma_f32_16x16x4(v2f a, v2f b, v8f c) {
  return __builtin_amdgcn_wmma_f32_16x16x4_f32(false, a, false, b, (short)0, c, false, false);
}

// v_rcp_f32-based sigmoid: avoids the IEEE v_div_scale/fixup slow path
__device__ __forceinline__ float fsig(float x) {
  return __builtin_amdgcn_rcpf(1.0f + __expf(-x));
}

#define SILU_C  1.67652f        // 1/sqrt(E[silu(z)^2]), z~N(0,1)
#define CC_S    0.3826834324f   // sin(pi/8)
#define CC_X    0.9238795325f   // cos(pi/8)
#define N800    0.03535533906f  // 1/sqrt(800)
#define INV_S10 0.31622776601f  // 1/sqrt(10)
#define INV_S64 0.125f          // 1/sqrt(64)
#define INV_S20 0.22360679775f  // 1/sqrt(20)
#define INV_NN  0.25f           // 1/sqrt(16)

// ws layout (floats): [0..19]=A_sc  [32..51]=A_l1  [64..463]=B2[u*20+c]
// [512..531]=wo  [1024..1024+N)=garr  then r[N]  then T[N*20]

// ---- fold the rank-1 s_x through the CG tensors --------------------------
__global__ void k_prep(const float* __restrict__ w_tp, const float* __restrict__ sc_ss,
                       const float* __restrict__ l1_ss, const float* __restrict__ l2_ss,
                       const float* __restrict__ w_out, float* __restrict__ ws) {
  __shared__ float t[20];
  const int tid = threadIdx.x;
  if (tid < 20) t[tid] = w_tp[tid];
  __syncthreads();
  for (int c = tid; c < 20; c += blockDim.x) {
    float s = 0.f, s2 = 0.f;
    for (int u = 0; u < 20; ++u)
      for (int v = 0; v < 20; ++v) {
        s  += sc_ss[(u*20 + v)*40 + c] * t[u] * t[v];
        s2 += l1_ss[(u*20 + v)*20 + c] * t[u] * t[v];
      }
    ws[c]       = s;
    ws[32 + c]  = s2;
    ws[512 + c] = w_out[c] * INV_S20;
  }
  for (int idx = tid; idx < 400; idx += blockDim.x) {
    const int u = idx / 20, c = idx % 20;
    float s = 0.f;
    for (int v = 0; v < 20; ++v) s += l2_ss[(u*20 + v)*40 + c] * t[v];
    ws[64 + idx] = s;
  }
}

// ---- garr[j] = emb[nf[src0[j]]]  (j < N < E0 so src[j] = ei[j]) ----------
__global__ void k_garr(const int* __restrict__ nf, const int* __restrict__ ei,
                       const float* __restrict__ emb, float* __restrict__ garr, int N) {
  const int j = blockIdx.x * blockDim.x + threadIdx.x;
  if (j < N) garr[j] = emb[nf[ei[j]]];
}

__global__ void k_zero(float* __restrict__ T, long long nT, float* __restrict__ out, int nOut) {
  const long long total = nT + nOut;
  for (long long i = (long long)blockIdx.x * blockDim.x + threadIdx.x; i < total;
       i += (long long)gridDim.x * blockDim.x) {
    if (i < nT) T[i] = 0.f; else out[i - nT] = 0.f;
  }
}

// ---- edge pass: RBF -> 10->64->64->20 MLP via V_WMMA_F32_16X16X4_F32 -----
// 8 waves/block, 16 edges/wave. Weights pre-scaled in LDS; per-wave H tile
// at stride 65 floats (bank = (row+col) mod 64 -> conflict free A loads).
__global__ __launch_bounds__(256) void k_edge(const int* __restrict__ ei,
                                              const float* __restrict__ pos,
                                              const float* __restrict__ garr,
                                              const float* __restrict__ rw0,
                                              const float* __restrict__ rw1,
                                              const float* __restrict__ rw2,
                                              float* __restrict__ T, int E0, int Etot) {
  __shared__ float sW0[12 * 64];   // rw0/sqrt(10), K padded 10->12
  __shared__ float sW1[64 * 64];   // rw1/sqrt(64)
  __shared__ float sW2[64 * 32];   // rw2[:, :20]/sqrt(64), N padded 20->32
  __shared__ float sH[8][16 * 65]; // per-wave activations (X -> h1 -> h2 -> w20)
  __shared__ int   sDst[8][16];
  __shared__ float sQ[8][16];

  const int tid = threadIdx.x;
  for (int i = tid; i < 12 * 64; i += 256) {
    const int k = i >> 6, n = i & 63;
    sW0[i] = (k < 10) ? rw0[k * 64 + n] * INV_S10 : 0.f;
  }
  for (int i = tid; i < 64 * 64; i += 256) sW1[i] = rw1[i] * INV_S64;
  for (int i = tid; i < 64 * 32; i += 256) {
    const int k = i >> 5, n = i & 31;
    sW2[i] = (n < 20) ? rw2[k * 40 + n] * INV_S64 : 0.f;
  }
  __syncthreads();

  const int w = tid >> 5, lane = tid & 31;
  const int hi = lane >> 4, ln = lane & 15;
  const int e0 = (blockIdx.x * 8 + w) * 16;
  if (e0 >= Etot) return;              // wave-uniform: EXEC stays all-ones for WMMA
  float* Hw = sH[w];

  if (hi == 0) {                       // lanes 0..15: one edge each
    const int m = ln;
    const int e = e0 + m;
    int dd = 0; float q = 0.f, dval = 0.f;
    const bool valid = (e < Etot);
    if (valid) {
      const int s = ei[e];                                 // src = ei_flat[e]
      dd = (e < E0) ? ei[e + E0] : ei[e - E0];             // dst = ei_flat[e +- E0]
      const float dx = pos[3*dd+0] - pos[3*s+0];
      const float dy = pos[3*dd+1] - pos[3*s+1];
      const float dz = pos[3*dd+2] - pos[3*s+2];
      dval = sqrtf(dx*dx + dy*dy + dz*dz);
      const float g = garr[s];
      q = g * g;                                           // q[src[e]]
    }
    float* Hr = &Hw[m * 65];
    #pragma unroll
    for (int k = 0; k < 10; ++k) {                         // RBF, width = 5/9
      const float d = dval - (5.0f / 9.0f) * (float)k;
      Hr[k] = valid ? __expf(-d * d * 3.24f) : 0.f;
    }
    Hr[10] = 0.f; Hr[11] = 0.f;
    sDst[w][m] = dd; sQ[w][m] = q;
  }
  __builtin_amdgcn_wave_barrier();

  const v8f z8 = {0, 0, 0, 0, 0, 0, 0, 0};

  // ---- layer 1: X[16x12] @ W0[12x64] ----
  v8f a1[4] = {z8, z8, z8, z8};
  #pragma unroll
  for (int k0 = 0; k0 < 12; k0 += 4) {
    const int ka = k0 + 2 * hi;                 // f32 A/B: VGPR0 holds K=ka, VGPR1 K=ka+1
    v2f a; a.x = Hw[ln * 65 + ka]; a.y = Hw[ln * 65 + ka + 1];
    #pragma unroll
    for (int t = 0; t < 4; ++t) {
      v2f b; b.x = sW0[ka * 64 + t * 16 + ln]; b.y = sW0[(ka + 1) * 64 + t * 16 + ln];
      a1[t] = wmma_f32_16x16x4(a, b, a1[t]);
    }
  }
  __builtin_amdgcn_wave_barrier();
  #pragma unroll
  for (int t = 0; t < 4; ++t)
    #pragma unroll
    for (int r = 0; r < 8; ++r) {               // D: VGPR r = rows r / r+8 per half-wave
      const float x = a1[t][r];
      Hw[(r + 8 * hi) * 65 + t * 16 + ln] = SILU_C * x * fsig(x);
    }
  __builtin_amdgcn_wave_barrier();

  // ---- layer 2: [16x64] @ W1[64x64], fully unrolled (64 WMMA) ----
  v8f a2[4] = {z8, z8, z8, z8};
  #pragma unroll
  for (int k0 = 0; k0 < 64; k0 += 4) {
    const int ka = k0 + 2 * hi;
    v2f a; a.x = Hw[ln * 65 + ka]; a.y = Hw[ln * 65 + ka + 1];
    #pragma unroll
    for (int t = 0; t < 4; ++t) {
      v2f b; b.x = sW1[ka * 64 + t * 16 + ln]; b.y = sW1[(ka + 1) * 64 + t * 16 + ln];
      a2[t] = wmma_f32_16x16x4(a, b, a2[t]);
    }
  }
  __builtin_amdgcn_wave_barrier();
  #pragma unroll
  for (int t = 0; t < 4; ++t)
    #pragma unroll
    for (int r = 0; r < 8; ++r) {
      const float x = a2[t][r];
      Hw[(r + 8 * hi) * 65 + t * 16 + ln] = SILU_C * x * fsig(x);
    }
  __builtin_amdgcn_wave_barrier();

  // ---- layer 3: [16x64] @ W2[64x32] (only cols 0..19 live), unrolled (32 WMMA) ----
  v8f a3[2] = {z8, z8};
  #pragma unroll
  for (int k0 = 0; k0 < 64; k0 += 4) {
    const int ka = k0 + 2 * hi;
    v2f a; a.x = Hw[ln * 65 + ka]; a.y = Hw[ln * 65 + ka + 1];
    #pragma unroll
    for (int t = 0; t < 2; ++t) {
      v2f b; b.x = sW2[ka * 32 + t * 16 + ln]; b.y = sW2[(ka + 1) * 32 + t * 16 + ln];
      a3[t] = wmma_f32_16x16x4(a, b, a3[t]);
    }
  }
  __builtin_amdgcn_wave_barrier();
  #pragma unroll
  for (int t = 0; t < 2; ++t)
    #pragma unroll
    for (int r = 0; r < 8; ++r) {
      const int col = t * 16 + ln;
      if (col < 20) Hw[(r + 8 * hi) * 65 + col] = a3[t][r];
    }
  __builtin_amdgcn_wave_barrier();

  // ---- scatter: T[dst, c] += q_src * w20[c]; lane handles 10 of 20 cols ----
  {
    const int m = ln;
    const int e = e0 + m;
    if (e < Etot) {
      const int dd = sDst[w][m];
      const float q = sQ[w][m];
      float* Trow = T + (long long)dd * 20;
      const int cb = hi * 10;
      #pragma unroll
      for (int c = 0; c < 10; ++c)
        atomicAdd(&Trow[cb + c], q * Hw[m * 65 + cb + c]);
    }
  }
}

// ---- node pass: s_m -> 20x20 matvec with B2 -> silu -> dot w_out ---------
__global__ __launch_bounds__(256) void k_node(const float* __restrict__ ws,
                                              const float* __restrict__ T,
                                              const float* __restrict__ garr,
                                              float* __restrict__ rOut, int N) {
  __shared__ float sAsc[20], sAl1[20], sB2[400], sWo[20];
  const int tid = threadIdx.x;
  if (tid < 20) { sAsc[tid] = ws[tid]; sAl1[tid] = ws[32 + tid]; sWo[tid] = ws[512 + tid]; }
  for (int i = tid; i < 400; i += blockDim.x) sB2[i] = ws[64 + i];
  __syncthreads();
  const int j = blockIdx.x * blockDim.x + tid;
  if (j >= N) return;
  float sm[20];
  #pragma unroll
  for (int u = 0; u < 20; ++u)
    sm[u] = INV_NN * N800 * sAl1[u] * T[(long long)j * 20 + u];
  const float g = garr[j];
  const float q = g * g;
  float acc = 0.f;
  #pragma unroll 4
  for (int c = 0; c < 20; ++c) {
    float s = 0.f;
    #pragma unroll
    for (int u = 0; u < 20; ++u) s += sm[u] * sB2[u * 20 + c];
    const float so = CC_S * (q * sAsc[c] * N800) + CC_X * (N800 * g * s);
    acc += (SILU_C * so * fsig(so)) * sWo[c];
  }
  rOut[j] = acc;
}

// ---- out[b] += r[src[e]] for every edge; LDS-binned then global atomics --
__global__ __launch_bounds__(256) void k_out(const int* __restrict__ ei,
                                             const int* __restrict__ batch,
                                             const float* __restrict__ r,
                                             float* __restrict__ out,
                                             long long nIdx, int B) {
  __shared__ float lacc[256];
  const int tid = threadIdx.x;
  lacc[tid] = 0.f;
  __syncthreads();
  for (long long i = (long long)blockIdx.x * blockDim.x + tid; i < nIdx;
       i += (long long)gridDim.x * blockDim.x) {
    const int j = ei[i];                 // src[e] over both directions = ei_flat
    atomicAdd(&lacc[batch[j] & 255], r[j]);
  }
  __syncthreads();
  if (tid < B) atomicAdd(&out[tid], lacc[tid]);
}

extern "C" void kernel_launch(void* const* d_in, const int* in_sizes, int n_in,
                              void* d_out, int out_size, void* d_ws, size_t ws_size,
                              hipStream_t stream) {
  const int*   nf    = (const int*)  d_in[0];
  const float* pos   = (const float*)d_in[1];
  const int*   ei    = (const int*)  d_in[2];
  const int*   batch = (const int*)  d_in[3];
  const float* emb   = (const float*)d_in[4];
  const float* w_tp  = (const float*)d_in[5];
  const float* rw0   = (const float*)d_in[6];
  const float* rw1   = (const float*)d_in[7];
  const float* rw2   = (const float*)d_in[8];
  const float* sc_ss = (const float*)d_in[9];
  const float* l1_ss = (const float*)d_in[14];
  const float* l2_ss = (const float*)d_in[18];
  const float* w_out = (const float*)d_in[23];
  // d_in[10..13], [15..17], [19..22] (vv0/sv/vs/vv2 tensors) multiply zero vectors.

  float* out = (float*)d_out;
  float* ws  = (float*)d_ws;

  const int N    = in_sizes[0];
  const int E0   = in_sizes[2] / 2;
  const int Etot = 2 * E0;

  float* garr = ws + 1024;
  float* rbuf = garr + N;
  float* T    = rbuf + N;

  k_prep<<<1, 256, 0, stream>>>(w_tp, sc_ss, l1_ss, l2_ss, w_out, ws);
  k_garr<<<(N + 255) / 256, 256, 0, stream>>>(nf, ei, emb, garr, N);

  const long long nT = (long long)N * 20;
  const long long zTot = nT + out_size;
  k_zero<<<(int)((zTot + 255) / 256), 256, 0, stream>>>(T, nT, out, out_size);

  k_edge<<<(Etot + 127) / 128, 256, 0, stream>>>(ei, pos, garr, rw0, rw1, rw2, T, E0, Etot);
  k_node<<<(N + 255) / 256, 256, 0, stream>>>(ws, T, garr, rbuf, N);
  k_out<<<1024, 256, 0, stream>>>(ei, batch, rbuf, out, (long long)Etot, out_size);
}